// DetectionOutput_43310450213512
// MI455X (gfx1250) — compile-verified
//
#include <hip/hip_runtime.h>
#include <stdint.h>

// amdgpu-toolchain (clang-23 + therock headers) ships the TDM header and uses
// the 6-arg tensor_load_to_lds builtin; ROCm 7.2 uses the 5-arg form.
#if __has_include(<hip/amd_detail/amd_gfx1250_TDM.h>)
#define TDM_SIX_ARG 1
#endif

#define NBINS     4096      // score histogram bins (scores uniform in [0,1])
#define CHUNK     2048      // priors per TDM tile (2*CHUNK floats = 16KB LDS)
#define CAND_CAP  1024      // per-image candidate buffer capacity
#define TOPK_     200
#define CONF_TH   0.01f
#define NMS_TH    0.45f

typedef unsigned int u32x4 __attribute__((ext_vector_type(4)));
typedef int          i32x8 __attribute__((ext_vector_type(8)));
typedef int          i32x4 __attribute__((ext_vector_type(4)));

__device__ __forceinline__ uint32_t lds_byte_off(const void* p) {
  // Generic (flat) LDS pointers carry the LDS byte address in the low 32 bits.
  return (uint32_t)(uintptr_t)p;
}

__device__ __forceinline__ int score_bin(float s) {
  int bn = (int)(s * (float)NBINS);
  return bn < 0 ? 0 : (bn >= NBINS ? NBINS - 1 : bn);
}

// Issue a 1-D TDM copy: tile_elems 4-byte elements from gptr to the LDS byte address in lds_byte.
// tensor_elems bounds the tensor (reads past it return zero -> harmless tail).
__device__ __forceinline__ void tdm_load_1d(uint32_t lds_byte, const void* gptr,
                                            uint32_t tensor_elems,
                                            uint32_t tile_elems) {
  uint64_t ga = (uint64_t)(uintptr_t)gptr;
  u32x4 g0;
  g0[0] = 1u;                                                // count=1, user D#
  g0[1] = lds_byte;                                          // lds_addr
  g0[2] = (uint32_t)ga;                                      // global_addr lo
  g0[3] = ((uint32_t)(ga >> 32) & 0x01FFFFFFu) | (2u << 30); // addr hi | type=2
  i32x8 g1;
  g1[0] = (int)(2u << 16);                                   // data_size=4B, wg_mask=0
  g1[1] = (int)((tensor_elems & 0xFFFFu) << 16);             // tensor_dim0 lo16
  g1[2] = (int)((tensor_elems >> 16) & 0xFFFFu) | (1 << 16); // dim0 hi16 | tensor_dim1=1
  g1[3] = (int)((tile_elems & 0xFFFFu) << 16);               // tile_dim0
  g1[4] = 0;                                                 // tile_dim1=0, tile_dim2=0
  g1[5] = (int)tensor_elems;                                 // dim0 stride (unused, 1-D)
  g1[6] = 0;
  g1[7] = 0;
  i32x4 z4 = {0, 0, 0, 0};
#ifdef TDM_SIX_ARG
  i32x8 z8 = {0, 0, 0, 0, 0, 0, 0, 0};
  __builtin_amdgcn_tensor_load_to_lds(g0, g1, z4, z4, z8, 0);
#else
  __builtin_amdgcn_tensor_load_to_lds(g0, g1, z4, z4, 0);
#endif
}

// ---------------------------------------------------------------------------
// Kernel 0: zero output / histograms / counters (re-run every call so graph
// replays are deterministic).
__global__ void k_init(float* out, int out_n, unsigned* hist, int hist_n,
                       unsigned* counts, int cnt_n) {
  int i = blockIdx.x * blockDim.x + threadIdx.x;
  int stride = gridDim.x * blockDim.x;
  for (int t = i; t < out_n; t += stride) out[t] = 0.0f;
  for (int t = i; t < hist_n; t += stride) hist[t] = 0u;
  for (int t = i; t < cnt_n; t += stride) counts[t] = 0u;
}

// ---------------------------------------------------------------------------
// Kernel 1: stream conf[b, :, :] through LDS with double-buffered TDM and
// histogram class-1 scores above the confidence threshold.
__global__ __launch_bounds__(256)
void k_hist(const float* __restrict__ conf, unsigned* __restrict__ hist,
            int N, int nchunks) {
  __shared__ float    sbuf[2][2 * CHUNK];  // 2 x 16KB tiles
  __shared__ unsigned shist[NBINS];        // 16KB histogram
  const int b   = blockIdx.x;
  const int tid = threadIdx.x;
  for (int i = tid; i < NBINS; i += 256) shist[i] = 0u;

  const float* base = conf + (size_t)b * N * 2;
  const int nblk = gridDim.y;
  const int c0   = blockIdx.y;

  if (tid < 32 && c0 < nchunks) {        // wave 0 issues the prologue tile
    int n0 = c0 * CHUNK;
    tdm_load_1d(lds_byte_off(&sbuf[0][0]), base + (size_t)n0 * 2,
                (uint32_t)(2 * (N - n0)), 2 * CHUNK);
  }

  int bufi = 0;
  for (int c = c0; c < nchunks; c += nblk) {
    int cn = c + nblk;
    if (tid < 32) {
      if (cn < nchunks) {                // prefetch next tile, keep 1 in flight
        int n0 = cn * CHUNK;
        tdm_load_1d(lds_byte_off(&sbuf[bufi ^ 1][0]), base + (size_t)n0 * 2,
                    (uint32_t)(2 * (N - n0)), 2 * CHUNK);
        __builtin_amdgcn_s_wait_tensorcnt(1);
      } else {
        __builtin_amdgcn_s_wait_tensorcnt(0);
      }
    }
    __syncthreads();                     // tile bufi is ready for all waves
    for (int t = tid; t < CHUNK; t += 256) {
      float s = sbuf[bufi][2 * t + 1];   // conf[:,1]; OOB tail is zero
      if (s > CONF_TH) atomicAdd(&shist[score_bin(s)], 1u);
    }
    __syncthreads();                     // done reading before tile is reused
    bufi ^= 1;
  }

  unsigned* gh = hist + (size_t)b * NBINS;
  for (int i = tid; i < NBINS; i += 256) {
    unsigned v = shist[i];
    if (v) atomicAdd(&gh[i], v);
  }
}

// ---------------------------------------------------------------------------
// Kernel 2: per image, find largest bin c with suffix-count >= min(200,total).
__global__ __launch_bounds__(256)
void k_cutoff(const unsigned* __restrict__ hist, int* __restrict__ cutoff) {
  __shared__ unsigned part[256];
  const int b = blockIdx.x;
  const int tid = threadIdx.x;
  const unsigned* h = hist + (size_t)b * NBINS;
  const int seg = NBINS / 256;  // 16 bins per thread
  unsigned s = 0;
  for (int i = 0; i < seg; ++i) s += h[tid * seg + i];
  part[tid] = s;
  __syncthreads();
  if (tid == 0) {
    unsigned total = 0;
    for (int i = 0; i < 256; ++i) total += part[i];
    unsigned need = total < (unsigned)TOPK_ ? total : (unsigned)TOPK_;
    int c = NBINS;  // "no candidates"
    if (need > 0) {
      unsigned acc = 0;
      int t = 255;
      for (; t >= 0; --t) {
        if (acc + part[t] >= need) break;
        acc += part[t];
      }
      c = t * seg;
      for (int bb = t * seg + seg - 1; bb >= t * seg; --bb) {
        acc += h[bb];
        if (acc >= need) { c = bb; break; }
      }
    }
    cutoff[b] = c;
  }
}

// ---------------------------------------------------------------------------
// Kernel 3: second TDM-streamed sweep — compact (score, idx) of everything in
// bins >= cutoff (a small superset of the exact top-200).
__global__ __launch_bounds__(256)
void k_compact(const float* __restrict__ conf, const int* __restrict__ cutoff,
               unsigned* __restrict__ cand_count, float* __restrict__ cand_s,
               int* __restrict__ cand_i, int N, int nchunks) {
  __shared__ float sbuf[2][2 * CHUNK];   // 2 x 16KB tiles
  const int b   = blockIdx.x;
  const int tid = threadIdx.x;
  const int cut = cutoff[b];

  const float* base = conf + (size_t)b * N * 2;
  const int nblk = gridDim.y;
  const int c0   = blockIdx.y;

  if (tid < 32 && c0 < nchunks) {        // wave 0 issues the prologue tile
    int n0 = c0 * CHUNK;
    tdm_load_1d(lds_byte_off(&sbuf[0][0]), base + (size_t)n0 * 2,
                (uint32_t)(2 * (N - n0)), 2 * CHUNK);
  }

  int bufi = 0;
  for (int c = c0; c < nchunks; c += nblk) {
    int cn = c + nblk;
    if (tid < 32) {
      if (cn < nchunks) {
        int n0 = cn * CHUNK;
        tdm_load_1d(lds_byte_off(&sbuf[bufi ^ 1][0]), base + (size_t)n0 * 2,
                    (uint32_t)(2 * (N - n0)), 2 * CHUNK);
        __builtin_amdgcn_s_wait_tensorcnt(1);
      } else {
        __builtin_amdgcn_s_wait_tensorcnt(0);
      }
    }
    __syncthreads();
    const int nbase = c * CHUNK;
    for (int t = tid; t < CHUNK; t += 256) {
      float s = sbuf[bufi][2 * t + 1];   // OOB tail is zero -> fails threshold
      if (s > CONF_TH && score_bin(s) >= cut) {
        unsigned pos = atomicAdd(&cand_count[b], 1u);
        if (pos < CAND_CAP) {
          cand_s[(size_t)b * CAND_CAP + pos] = s;
          cand_i[(size_t)b * CAND_CAP + pos] = nbase + t;
        }
      }
    }
    __syncthreads();
    bufi ^= 1;
  }
}

// ---------------------------------------------------------------------------
// Kernel 4: per image — exact sort of candidates (score desc, idx asc like
// lax.top_k), decode only the 200 winners, greedy NMS matching the lax.scan.
__global__ __launch_bounds__(256)
void k_final(const float* __restrict__ loc, const float* __restrict__ priors,
             const float* __restrict__ cand_s, const int* __restrict__ cand_i,
             const unsigned* __restrict__ cand_count, float* __restrict__ out,
             int N) {
  __shared__ float cs[CAND_CAP];
  __shared__ int   ci[CAND_CAP];
  __shared__ float ts[TOPK_];
  __shared__ float bx0[TOPK_], by0[TOPK_], bx1[TOPK_], by1[TOPK_];
  __shared__ int   act[TOPK_];
  __shared__ int   pick;
  const int b = blockIdx.x;
  const int tid = threadIdx.x;

  int m = (int)cand_count[b];
  if (m > CAND_CAP) m = CAND_CAP;
  for (int t = tid; t < CAND_CAP; t += 256) {
    if (t < m) {
      cs[t] = cand_s[(size_t)b * CAND_CAP + t];
      ci[t] = cand_i[(size_t)b * CAND_CAP + t];
    } else {
      cs[t] = -1.0f;
      ci[t] = 0x7FFFFFFF;
    }
  }
  __syncthreads();

  // Bitonic sort, order: score descending, ascending index on ties.
  for (int k = 2; k <= CAND_CAP; k <<= 1) {
    for (int j = k >> 1; j > 0; j >>= 1) {
      for (int t = tid; t < CAND_CAP; t += 256) {
        int x = t ^ j;
        if (x > t) {
          float s1 = cs[t], s2 = cs[x];
          int   i1 = ci[t], i2 = ci[x];
          bool before = (s1 > s2) || (s1 == s2 && i1 < i2);
          bool up = ((t & k) == 0);
          if (up ? !before : before) {
            cs[t] = s2; cs[x] = s1;
            ci[t] = i2; ci[x] = i1;
          }
        }
      }
      __syncthreads();
    }
  }

  // Decode only the top-200 boxes (variances 0.1 / 0.2).
  if (tid < TOPK_) {
    bool v = (tid < m);
    float sc = v ? cs[tid] : 0.0f;
    int   id = v ? ci[tid] : 0;
    float4 l = ((const float4*)loc)[(size_t)b * N + id];
    float4 p = ((const float4*)priors)[id];
    float cx = p.x + l.x * 0.1f * p.z;
    float cy = p.y + l.y * 0.1f * p.w;
    float w  = p.z * expf(l.z * 0.2f);
    float h  = p.w * expf(l.w * 0.2f);
    ts[tid]  = sc;
    bx0[tid] = cx - 0.5f * w;
    by0[tid] = cy - 0.5f * h;
    bx1[tid] = cx + 0.5f * w;
    by1[tid] = cy + 0.5f * h;
    act[tid] = (v && sc > CONF_TH) ? 1 : 0;
  }
  __syncthreads();

  // Greedy NMS: on a descending-sorted list, argmax == first active entry.
  float* orow = out + ((size_t)(b * 2 + 1) * TOPK_) * 5;  // class 1 rows
  for (int step = 0; step < TOPK_; ++step) {
    if (tid == 0) {
      int sel = -1;
      for (int t = 0; t < TOPK_; ++t)
        if (act[t]) { sel = t; break; }
      pick = sel;
    }
    __syncthreads();
    int i = pick;
    if (i < 0) break;  // remaining rows stay zero, matching keep=False rows
    if (tid < TOPK_) {
      float lt0 = fmaxf(bx0[i], bx0[tid]);
      float lt1 = fmaxf(by0[i], by0[tid]);
      float rb0 = fminf(bx1[i], bx1[tid]);
      float rb1 = fminf(by1[i], by1[tid]);
      float w = fmaxf(rb0 - lt0, 0.0f);
      float h = fmaxf(rb1 - lt1, 0.0f);
      float inter = w * h;
      float a1 = (bx1[i] - bx0[i]) * (by1[i] - by0[i]);
      float a2 = (bx1[tid] - bx0[tid]) * (by1[tid] - by0[tid]);
      float iou = inter / (a1 + a2 - inter);
      if (act[tid] && !(iou <= NMS_TH)) act[tid] = 0;  // NaN => suppressed
    }
    __syncthreads();
    if (tid == 0) {
      act[i] = 0;
      float* r = orow + (size_t)step * 5;
      r[0] = ts[i]; r[1] = bx0[i]; r[2] = by0[i]; r[3] = bx1[i]; r[4] = by1[i];
    }
    __syncthreads();
  }
}

// ---------------------------------------------------------------------------
extern "C" void kernel_launch(void* const* d_in, const int* in_sizes, int n_in,
                              void* d_out, int out_size, void* d_ws,
                              size_t ws_size, hipStream_t stream) {
  const float* loc    = (const float*)d_in[0];  // (B, N, 4) f32
  const float* conf   = (const float*)d_in[1];  // (B, N, 2) f32
  const float* priors = (const float*)d_in[2];  // (N, 4)    f32
  float* out = (float*)d_out;                   // (B, 2, 200, 5) f32

  const int N = in_sizes[2] / 4;
  const int B = (N > 0) ? in_sizes[0] / (N * 4) : 0;
  if (B <= 0 || N <= 0) return;

  // Workspace layout (~770KB for B=32): hist | counts | cutoff | cand_s | cand_i
  char* ws = (char*)d_ws;
  unsigned* hist = (unsigned*)ws;
  size_t off = (size_t)B * NBINS * sizeof(unsigned);
  unsigned* cand_count = (unsigned*)(ws + off); off += (size_t)B * sizeof(unsigned);
  int*      cutoff     = (int*)(ws + off);      off += (size_t)B * sizeof(int);
  float*    cand_s     = (float*)(ws + off);    off += (size_t)B * CAND_CAP * sizeof(float);
  int*      cand_i     = (int*)(ws + off);

  const int nchunks = (N + CHUNK - 1) / CHUNK;

  k_init<<<256, 256, 0, stream>>>(out, out_size, hist, B * NBINS, cand_count, B);
  dim3 sgrid((unsigned)B, 25);
  k_hist<<<sgrid, 256, 0, stream>>>(conf, hist, N, nchunks);
  k_cutoff<<<B, 256, 0, stream>>>(hist, cutoff);
  k_compact<<<sgrid, 256, 0, stream>>>(conf, cutoff, cand_count, cand_s, cand_i,
                                       N, nchunks);
  k_final<<<B, 256, 0, stream>>>(loc, priors, cand_s, cand_i, cand_count, out, N);
}